// RPNLoss_44478681317476
// MI455X (gfx1250) — compile-verified
//
#include <hip/hip_runtime.h>
#include <math.h>

typedef __attribute__((ext_vector_type(2))) float v2f;
typedef __attribute__((ext_vector_type(8))) float v8f;

namespace {
constexpr int BATCH = 16;
constexpr int H     = 352;
constexpr int W     = 1216;          // divisible by 4 -> clean float4 tiling
constexpr int NIMG  = H * W;
constexpr int TPB   = 256;
constexpr int PXT   = 4;                              // pixels per thread (float4)
constexpr int XSPAN = TPB * PXT;                      // 1024 pixels per block in x
constexpr int XB    = (W + XSPAN - 1) / XSPAN;        // 2
constexpr int STRIP = 16;
constexpr int YB    = (H + STRIP - 1) / STRIP;        // 22
constexpr int BPI   = XB * YB;                        // 44 blocks per image
}

#define SORT2(a, b) { float _t = fminf(a, b); b = fmaxf(a, b); a = _t; }

__device__ __forceinline__ float median9(float v0, float v1, float v2,
                                         float v3, float v4, float v5,
                                         float v6, float v7, float v8) {
  // 19-op median-of-9 sorting network (exact median; == lower median for 9).
  SORT2(v1, v2); SORT2(v4, v5); SORT2(v7, v8);
  SORT2(v0, v1); SORT2(v3, v4); SORT2(v6, v7);
  SORT2(v1, v2); SORT2(v4, v5); SORT2(v7, v8);
  SORT2(v0, v3); SORT2(v5, v8); SORT2(v4, v7);
  SORT2(v3, v6); SORT2(v1, v4); SORT2(v2, v5);
  SORT2(v4, v7); SORT2(v4, v2); SORT2(v6, v4);
  SORT2(v4, v2);
  return v4;
}

// Load a 6-wide window [x4-1 .. x4+4] of row y (zero-padded) using one b128
// load for the 4 owned pixels plus two edge b32 loads. x4 is 4-aligned.
__device__ __forceinline__ void load_row6(const float* __restrict__ img, int y,
                                          int x4, float* __restrict__ r) {
  if (y < 0 || y >= H) {
#pragma unroll
    for (int i = 0; i < 6; ++i) r[i] = 0.0f;
    return;
  }
  const float* row = img + y * W;
  const float4 c = *(const float4*)(row + x4);
  r[1] = c.x; r[2] = c.y; r[3] = c.z; r[4] = c.w;
  r[0] = (x4 > 0)     ? row[x4 - 1] : 0.0f;
  r[5] = (x4 + 4 < W) ? row[x4 + 4] : 0.0f;
}

// Fixed-order block reduction: wave butterfly (32 lanes), then thread 0 sums
// the per-wave results in a fixed order. Deterministic. Result valid in tid 0.
__device__ __forceinline__ float block_sum(float v, float* sred) {
  for (int off = 16; off >= 1; off >>= 1) v += __shfl_xor(v, off, 32);
  const int lane = threadIdx.x & 31;
  const int wid  = threadIdx.x >> 5;
  if (lane == 0) sred[wid] = v;
  __syncthreads();
  float r = 0.0f;
  if (threadIdx.x == 0) {
    const int nw = (int)(blockDim.x >> 5);
    for (int i = 0; i < nw; ++i) r += sred[i];
  }
  __syncthreads();
  return r;
}

// Sum n contiguous f32 values with ONE wave (all 32 lanes execute, EXEC all-1)
// via V_WMMA_F32_16X16X4_F32 with A == ones(16x4): every B element contributes
// exactly once to the accumulated column sums. Total = sum_n D[0][n] = sum of
// c[0] over lanes 0..15 = 0.5 * butterfly-sum over all 32 lanes.
__device__ __forceinline__ float wave_sum_wmma(const float* __restrict__ p, int n) {
  const int lane = (int)(threadIdx.x & 31);
  v8f c = {};
  v2f ones; ones.x = 1.0f; ones.y = 1.0f;
  for (int base = 0; base < n; base += 64) {
    const int i0 = base + lane;
    const int i1 = base + 32 + lane;
    v2f b;
    b.x = (i0 < n) ? p[i0] : 0.0f;
    b.y = (i1 < n) ? p[i1] : 0.0f;
    c = __builtin_amdgcn_wmma_f32_16x16x4_f32(false, ones, false, b,
                                              (short)0, c, false, false);
  }
  float v = c[0];
  for (int off = 16; off >= 1; off >>= 1) v += __shfl_xor(v, off, 32);
  return 0.5f * v;
}

// ---------------------------------------------------------------------------
// Pass 1: per-block partial sums of mask, mask*|pred_contrast|, mask*|tgt_contrast|
// Each thread owns 4 consecutive pixels and slides a 3-row window down a strip.
// ---------------------------------------------------------------------------
__global__ void __launch_bounds__(TPB)
partials_kernel(const float* __restrict__ pred, const float* __restrict__ tgt,
                const float* __restrict__ mask,
                float* __restrict__ pMask, float* __restrict__ pPred,
                float* __restrict__ pTgt) {
  __shared__ float sred[TPB / 32];
  const int b  = blockIdx.z;
  const int x4 = (blockIdx.x * TPB + (int)threadIdx.x) * PXT;
  const int y0 = blockIdx.y * STRIP;
  const int y1 = min(y0 + STRIP, H);
  const float* pi = pred + b * NIMG;
  const float* ti = tgt  + b * NIMG;
  const float* mi = mask + b * NIMG;

  float sM = 0.0f, sP = 0.0f, sT = 0.0f;
  if (x4 < W) {
    float p[3][6], t[3][6];
    load_row6(pi, y0 - 1, x4, p[0]);
    load_row6(pi, y0,     x4, p[1]);
    load_row6(ti, y0 - 1, x4, t[0]);
    load_row6(ti, y0,     x4, t[1]);
    for (int y = y0; y < y1; ++y) {
      load_row6(pi, y + 1, x4, p[2]);
      load_row6(ti, y + 1, x4, t[2]);
      if (y + 2 < H) __builtin_prefetch(pi + (y + 2) * W + x4, 0, 3);
      const float4 m4 = *(const float4*)(mi + y * W + x4);
      const float m[4] = {m4.x, m4.y, m4.z, m4.w};
#pragma unroll
      for (int j = 0; j < PXT; ++j) {
        const float mp = median9(p[0][j], p[0][j + 1], p[0][j + 2],
                                 p[1][j], p[1][j + 1], p[1][j + 2],
                                 p[2][j], p[2][j + 1], p[2][j + 2]);
        const float mt = median9(t[0][j], t[0][j + 1], t[0][j + 2],
                                 t[1][j], t[1][j + 1], t[1][j + 2],
                                 t[2][j], t[2][j + 1], t[2][j + 2]);
        sM += m[j];
        sP += m[j] * fabsf(p[1][j + 1] - mp);
        sT += m[j] * fabsf(t[1][j + 1] - mt);
      }
#pragma unroll
      for (int i = 0; i < 6; ++i) {
        p[0][i] = p[1][i]; p[1][i] = p[2][i];
        t[0][i] = t[1][i]; t[1][i] = t[2][i];
      }
    }
  }
  const float rM = block_sum(sM, sred);
  const float rP = block_sum(sP, sred);
  const float rT = block_sum(sT, sred);
  if (threadIdx.x == 0) {
    const int blk = b * BPI + (int)blockIdx.y * XB + (int)blockIdx.x;
    pMask[blk] = rM; pPred[blk] = rP; pTgt[blk] = rT;
  }
}

// ---------------------------------------------------------------------------
// Pass 2: per-image stats. One wave per image; WMMA reduction.
// ---------------------------------------------------------------------------
__global__ void __launch_bounds__(32)
stats_kernel(const float* __restrict__ pMask, const float* __restrict__ pPred,
             const float* __restrict__ pTgt,
             float* __restrict__ rp, float* __restrict__ rt,
             float* __restrict__ den) {
  const int b = blockIdx.x;
  const float sM = wave_sum_wmma(pMask + b * BPI, BPI);
  const float sP = wave_sum_wmma(pPred + b * BPI, BPI);
  const float sT = wave_sum_wmma(pTgt  + b * BPI, BPI);
  if (threadIdx.x == 0) {
    rp[b]  = sM / sP;   // 1/mean_pred
    rt[b]  = sM / sT;   // 1/mean_target
    den[b] = sM;
  }
}

// ---------------------------------------------------------------------------
// Pass 3: loss partials: sum |pc*rp - tc*rt| per block (inputs hit L2 now).
// ---------------------------------------------------------------------------
__global__ void __launch_bounds__(TPB)
loss_kernel(const float* __restrict__ pred, const float* __restrict__ tgt,
            const float* __restrict__ rp, const float* __restrict__ rt,
            float* __restrict__ pLoss) {
  __shared__ float sred[TPB / 32];
  const int b  = blockIdx.z;
  const int x4 = (blockIdx.x * TPB + (int)threadIdx.x) * PXT;
  const int y0 = blockIdx.y * STRIP;
  const int y1 = min(y0 + STRIP, H);
  const float* pi = pred + b * NIMG;
  const float* ti = tgt  + b * NIMG;
  const float srp = rp[b];
  const float srt = rt[b];

  float sL = 0.0f;
  if (x4 < W) {
    float p[3][6], t[3][6];
    load_row6(pi, y0 - 1, x4, p[0]);
    load_row6(pi, y0,     x4, p[1]);
    load_row6(ti, y0 - 1, x4, t[0]);
    load_row6(ti, y0,     x4, t[1]);
    for (int y = y0; y < y1; ++y) {
      load_row6(pi, y + 1, x4, p[2]);
      load_row6(ti, y + 1, x4, t[2]);
#pragma unroll
      for (int j = 0; j < PXT; ++j) {
        const float mp = median9(p[0][j], p[0][j + 1], p[0][j + 2],
                                 p[1][j], p[1][j + 1], p[1][j + 2],
                                 p[2][j], p[2][j + 1], p[2][j + 2]);
        const float mt = median9(t[0][j], t[0][j + 1], t[0][j + 2],
                                 t[1][j], t[1][j + 1], t[1][j + 2],
                                 t[2][j], t[2][j + 1], t[2][j + 2]);
        const float pc = p[1][j + 1] - mp;
        const float tc = t[1][j + 1] - mt;
        sL += fabsf(pc * srp - tc * srt);
      }
#pragma unroll
      for (int i = 0; i < 6; ++i) {
        p[0][i] = p[1][i]; p[1][i] = p[2][i];
        t[0][i] = t[1][i]; t[1][i] = t[2][i];
      }
    }
  }
  const float rL = block_sum(sL, sred);
  if (threadIdx.x == 0) {
    pLoss[b * BPI + (int)blockIdx.y * XB + (int)blockIdx.x] = rL;
  }
}

// ---------------------------------------------------------------------------
// Pass 4: final scalar: sum(loss) / sum(mask). One wave, WMMA reduction.
// ---------------------------------------------------------------------------
__global__ void __launch_bounds__(32)
final_kernel(const float* __restrict__ pLoss, const float* __restrict__ den,
             float* __restrict__ out) {
  const float lsum = wave_sum_wmma(pLoss, BATCH * BPI);
  const float msum = wave_sum_wmma(den, BATCH);
  if (threadIdx.x == 0) out[0] = lsum / msum;
}

extern "C" void kernel_launch(void* const* d_in, const int* in_sizes, int n_in,
                              void* d_out, int out_size, void* d_ws, size_t ws_size,
                              hipStream_t stream) {
  (void)in_sizes; (void)n_in; (void)out_size; (void)ws_size;
  const float* pred = (const float*)d_in[0];
  const float* tgt  = (const float*)d_in[1];
  const float* mask = (const float*)d_in[2];
  float* out = (float*)d_out;
  float* ws  = (float*)d_ws;

  float* pMask = ws;                       // BATCH*BPI = 704
  float* pPred = pMask + BATCH * BPI;      // 704
  float* pTgt  = pPred + BATCH * BPI;      // 704
  float* rp    = pTgt  + BATCH * BPI;      // 16
  float* rt    = rp + BATCH;               // 16
  float* den   = rt + BATCH;               // 16
  float* pLoss = den + BATCH;              // 704   (total ~11 KB)

  dim3 grid(XB, YB, BATCH);
  partials_kernel<<<grid, TPB, 0, stream>>>(pred, tgt, mask, pMask, pPred, pTgt);
  stats_kernel<<<BATCH, 32, 0, stream>>>(pMask, pPred, pTgt, rp, rt, den);
  loss_kernel<<<grid, TPB, 0, stream>>>(pred, tgt, rp, rt, pLoss);
  final_kernel<<<1, 32, 0, stream>>>(pLoss, den, out);
}